// AttentionCTCLoss_35304631173598
// MI455X (gfx1250) — compile-verified
//
#include <hip/hip_runtime.h>
#include <math.h>

#define BB   128
#define TQc  900
#define TKc  200
#define Sc   401           // 2*TKc + 1 lattice states
#define DPF  4             // prefetch ring depth
#define NEGF (-1.0e30f)
#define BLANK_LP (-1.0f)

// ---------- CDNA5 async global->LDS support (guarded; falls back to sync) ----
#if defined(__has_builtin)
#  if __has_builtin(__builtin_amdgcn_global_load_async_to_lds_b32)
#    define CTC_ASYNC 1
#  endif
#endif
#ifndef CTC_ASYNC
#  define CTC_ASYNC 0
#endif

typedef __attribute__((address_space(1))) int* gint_ptr;
typedef __attribute__((address_space(3))) int* lint_ptr;

template <int N>
__device__ __forceinline__ void async_wait_le() {
#if CTC_ASYNC
#  if defined(__has_builtin) && __has_builtin(__builtin_amdgcn_s_wait_asynccnt)
  __builtin_amdgcn_s_wait_asynccnt(N);
#  else
  asm volatile("s_wait_asynccnt %0" :: "i"(N) : "memory");
#  endif
#endif
}

// Fetch one attn row (TKc floats) + its lse scalar into LDS row buffer.
// Slots [0..TKc-1] = attn row, slot [TKc] = lse.
__device__ __forceinline__ void prefetch_row(const float* __restrict__ grow,
                                             const float* __restrict__ glse,
                                             float* ldst, int s) {
#if CTC_ASYNC
  if (s < TKc) {
    __builtin_amdgcn_global_load_async_to_lds_b32(
        (gint_ptr)(grow + s), (lint_ptr)(ldst + s), 0, 0);
  } else if (s == TKc) {
    __builtin_amdgcn_global_load_async_to_lds_b32(
        (gint_ptr)glse, (lint_ptr)(ldst + TKc), 0, 0);
  }
#else
  if (s < TKc)       ldst[s]   = grow[s];
  else if (s == TKc) ldst[TKc] = *glse;
#endif
}

__device__ __forceinline__ float logaddexp2f(float x, float y) {
  float mx = fmaxf(x, y);
  float mn = fminf(x, y);
  return mx + log1pf(expf(mn - mx));
}

// ---------------- Phase 1: lse[b,t] = logsumexp over padded 201-wide row -----
// One wave32 per (b,t) row.
__global__ void lse_kernel(const float* __restrict__ attn, float* __restrict__ lse) {
  const int gwave = (int)((blockIdx.x * blockDim.x + threadIdx.x) >> 5);
  const int lane  = (int)(threadIdx.x & 31);
  const int nrows = BB * TQc;
  if (gwave >= nrows) return;
  const float* row = attn + (size_t)gwave * TKc;

  float v[7];
  float m = -3.0e38f;
#pragma unroll
  for (int j = 0; j < 7; ++j) {
    int c = lane + 32 * j;              // padded class index 0..200
    float x;
    if (c == 0)         x = BLANK_LP;
    else if (c <= TKc)  x = row[c - 1];
    else                x = -3.0e38f;   // inert for max and sum
    v[j] = x;
    m = fmaxf(m, x);
  }
#pragma unroll
  for (int off = 16; off >= 1; off >>= 1)
    m = fmaxf(m, __shfl_xor(m, off, 32));

  float sum = 0.0f;
#pragma unroll
  for (int j = 0; j < 7; ++j)
    sum += expf(v[j] - m);              // exp(-3e38 - m) == 0
#pragma unroll
  for (int off = 16; off >= 1; off >>= 1)
    sum += __shfl_xor(sum, off, 32);

  if (lane == 0) lse[gwave] = m + logf(sum);
}

// ---------------- Phase 2: sequential alpha recursion, one block per batch ---
// One thread per lattice state; DPF-deep async row pipeline; alpha ping-pong
// in LDS; early exit at t = out_len-1 (fin is frozen after capture).
__global__ __launch_bounds__(416) void ctc_alpha_kernel(
    const float* __restrict__ attn, const float* __restrict__ lse,
    const int* __restrict__ in_lens, const int* __restrict__ out_lens,
    float* __restrict__ loss_out) {
  __shared__ float rowb[DPF][TKc + 8];  // ring: attn row + lse in slot TKc
  __shared__ float alp[2][Sc + 4];      // alpha with 2-slot NEG pad at front

  const int s = (int)threadIdx.x;       // lattice state (0..Sc-1 active)
  const int b = (int)blockIdx.x;
  const int in_len  = in_lens[b];
  const int out_len = out_lens[b];
  const float* arow = attn + (size_t)b * TQc * TKc;
  const float* lrow = lse  + (size_t)b * TQc;

  if (s < 2) { alp[0][s] = NEGF; alp[1][s] = NEGF; }

  // Prime the pipeline: rows 0..DPF-2
#pragma unroll
  for (int r = 0; r < DPF - 1; ++r)
    if (r < out_len)
      prefetch_row(arow + (size_t)r * TKc, lrow + r, &rowb[r][0], s);

  // ---- t = 0 (init step, same wait protocol) ----
  async_wait_le<DPF - 2>();             // row 0 landed (in-order completion)
  __syncthreads();
  if (out_len > DPF - 1)                // issue row DPF-1 into buffer DPF-1
    prefetch_row(arow + (size_t)(DPF - 1) * TKc, lrow + (DPF - 1),
                 &rowb[(DPF - 1) % DPF][0], s);
  {
    float l0 = rowb[0][TKc];
    float e = NEGF;
    if (s == 0)      e = BLANK_LP - l0;
    else if (s == 1) e = (0 < in_len) ? (rowb[0][0] - l0) : 0.0f;
    if (s < Sc) alp[0][s + 2] = e;
  }

  int cur = 0;
  for (int t = 1; t < out_len; ++t) {
    async_wait_le<DPF - 2>();           // my row-t ops retired (oldest first)
    __syncthreads();                    // everyone's row t landed; alpha visible;
                                        // buffer (t-1)%DPF fully consumed
    int nr = t + DPF - 1;               // refill the just-freed buffer
    if (nr < out_len)
      prefetch_row(arow + (size_t)nr * TKc, lrow + nr, &rowb[nr % DPF][0], s);

    const float* rv = rowb[t % DPF];
    if (s < Sc) {
      const float lset = rv[TKc];
      float e;
      if ((s & 1) == 0) {
        e = BLANK_LP - lset;                         // blank state
      } else {
        int k = (s - 1) >> 1;                        // attn column
        e = (k < in_len) ? (rv[k] - lset) : 0.0f;    // kmask zeroes logp
      }
      const float* ac = alp[cur];
      float a0 = ac[s + 2];
      float a1 = ac[s + 1];
      float a2 = ((s & 1) && s >= 2) ? ac[s] : NEGF; // skip transition
      float m  = fmaxf(fmaxf(a0, a1), a2);
      float se = expf(a0 - m) + expf(a1 - m) + expf(a2 - m);
      alp[cur ^ 1][s + 2] = m + logf(se) + e;
    }
    cur ^= 1;
  }

  __syncthreads();                      // publish final alpha (== fin)
  if (s == 0) {
    const float* fin = alp[cur];
    float hi = fin[2 * in_len + 2];     // state 2*in_len
    float lo = fin[2 * in_len + 1];     // state 2*in_len - 1
    float l = -logaddexp2f(hi, lo);
    if (l > 1e20f) l = 0.0f;
    loss_out[b] = l / (float)in_len;
  }
}

// ---------------- Phase 3: deterministic mean over batch ---------------------
__global__ void mean_kernel(const float* __restrict__ loss, float* __restrict__ out) {
  __shared__ float sm[BB];
  int t = (int)threadIdx.x;
  sm[t] = loss[t];
  __syncthreads();
  for (int off = BB / 2; off > 0; off >>= 1) {
    if (t < off) sm[t] += sm[t + off];
    __syncthreads();
  }
  if (t == 0) out[0] = sm[0] / (float)BB;
}

extern "C" void kernel_launch(void* const* d_in, const int* in_sizes, int n_in,
                              void* d_out, int out_size, void* d_ws, size_t ws_size,
                              hipStream_t stream) {
  const float* attn   = (const float*)d_in[0];
  const int* in_lens  = (const int*)d_in[1];
  const int* out_lens = (const int*)d_in[2];

  float* lse  = (float*)d_ws;                 // BB*TQc floats
  float* loss = lse + (size_t)BB * TQc;       // BB floats

  const int nrows = BB * TQc;                 // one wave32 per row
  const int wavesPerBlock = 8;                // 256 threads
  const int blocks1 = (nrows + wavesPerBlock - 1) / wavesPerBlock;

  lse_kernel<<<blocks1, 256, 0, stream>>>(attn, lse);
  ctc_alpha_kernel<<<BB, 416, 0, stream>>>(attn, lse, in_lens, out_lens, loss);
  mean_kernel<<<1, BB, 0, stream>>>(loss, (float*)d_out);
}